// NetRGCN_29137058136732
// MI455X (gfx1250) — compile-verified
//
#include <hip/hip_runtime.h>
#include <hip/hip_bf16.h>

// Problem constants
#define Bsz 32
#define Nn  2048
#define Ee  8192
#define Dd  512
#define Rr  8
#define VG  50000
#define VS  25000
#define KX  ((Rr + 1) * Dd)   // 4608

typedef float v2f __attribute__((ext_vector_type(2)));
typedef float v8f __attribute__((ext_vector_type(8)));

__device__ __forceinline__ v8f wmma4(v2f a, v2f b, v8f c) {
    // D(16x16,f32) = A(16x4,f32) * B(4x16,f32) + C
    return __builtin_amdgcn_wmma_f32_16x16x4_f32(
        /*neg_a=*/false, a, /*neg_b=*/false, b,
        /*c_mod=*/(short)0, c, /*reuse_a=*/false, /*reuse_b=*/false);
}

// ---------------------------------------------------------------------------
// K0: W_all[0] = root ; W_all[1+r] = sum_b comp[r,b]*basis[b]   (9 x 512 x 512)
// ---------------------------------------------------------------------------
__global__ void rgcn_build_weights(const float* __restrict__ root,
                                   const float* __restrict__ comp,
                                   const float* __restrict__ basis,
                                   float* __restrict__ Wall) {
    const int DD = Dd * Dd;
    int idx = blockIdx.x * blockDim.x + threadIdx.x;
    if (idx >= (Rr + 1) * DD) return;
    int slab = idx / DD;
    int rem  = idx - slab * DD;
    if (slab == 0) {
        Wall[idx] = root[rem];
    } else {
        int r = slab - 1;
        float s = 0.0f;
#pragma unroll
        for (int b = 0; b < Rr; ++b)
            s += comp[r * Rr + b] * basis[(size_t)b * DD + rem];
        Wall[idx] = s;
    }
}

// ---------------------------------------------------------------------------
// K1: deterministic aggregation of edges into node 0, per (batch, relation).
// One block per batch; all threads redundantly scan edges in fixed e-order so
// the float summation order is identical on every call (no atomics).
// Each thread owns 2 feature dims (256 threads * 2 = 512).
// ---------------------------------------------------------------------------
__global__ __launch_bounds__(256)
void rgcn_edge_agg(const int* __restrict__ ei, const int* __restrict__ et,
                   const float* __restrict__ x,
                   float* __restrict__ agg0, float* __restrict__ cnt0) {
    const int b   = blockIdx.x;
    const int tid = threadIdx.x;
    const int* src_p = ei + (size_t)b * 2 * Ee;
    const int* dst_p = src_p + Ee;
    const int* type_p = et + (size_t)b * Ee;

    for (int r = 0; r < Rr; ++r) {
        float a0 = 0.0f, a1 = 0.0f;
        int cnt = 0;
        for (int e = 0; e < Ee; ++e) {
            if (dst_p[e] == 0 && type_p[e] == r) {
                int src = src_p[e];
                const float* xs = x + ((size_t)b * Nn + src) * Dd;
                a0 += xs[tid];
                a1 += xs[tid + 256];
                ++cnt;
            }
        }
        float* ap = agg0 + ((size_t)b * Rr + r) * Dd;
        ap[tid]       = a0;
        ap[tid + 256] = a1;
        if (tid == 0) cnt0[b * Rr + r] = (float)cnt;
    }
}

// ---------------------------------------------------------------------------
// K2: build extended activation Aext(32 x 4608) = [ x[b,0,:], agg0/max(cnt,1) ]
// ---------------------------------------------------------------------------
__global__ void rgcn_build_aext(const float* __restrict__ x,
                                const float* __restrict__ agg0,
                                const float* __restrict__ cnt0,
                                float* __restrict__ Aext) {
    int t = blockIdx.x * blockDim.x + threadIdx.x;
    if (t >= Bsz * KX) return;
    int b = t / KX;
    int j = t - b * KX;
    float v;
    if (j < Dd) {
        v = x[((size_t)b * Nn) * Dd + j];            // node 0 features
    } else {
        int r = j / Dd - 1;
        int d = j & (Dd - 1);
        float c = cnt0[b * Rr + r];
        v = agg0[((size_t)b * Rr + r) * Dd + d] / fmaxf(c, 1.0f);
    }
    Aext[t] = v;
}

// ---------------------------------------------------------------------------
// K3: h0 = relu(Aext(32x4608) @ Wall(4608x512) + bias)   — WMMA f32 16x16x4
// grid = (Dd/16, 2), block = 32 (one wave, EXEC all ones)
// ---------------------------------------------------------------------------
__global__ __launch_bounds__(32)
void rgcn_conv_gemm(const float* __restrict__ Aext,
                    const float* __restrict__ Wall,
                    const float* __restrict__ bias,
                    float* __restrict__ h0) {
    const int lane = threadIdx.x;
    const int half = lane >> 4;
    const int lr   = lane & 15;
    const int n0   = blockIdx.x * 16;
    const int m0   = blockIdx.y * 16;
    v8f c = {0.f, 0.f, 0.f, 0.f, 0.f, 0.f, 0.f, 0.f};
    for (int k0 = 0; k0 < KX; k0 += 4) {
        const int ka = k0 + 2 * half;
        v2f a, b;
        a.x = Aext[(size_t)(m0 + lr) * KX + ka];
        a.y = Aext[(size_t)(m0 + lr) * KX + ka + 1];
        b.x = Wall[(size_t)ka * Dd + n0 + lr];
        b.y = Wall[(size_t)(ka + 1) * Dd + n0 + lr];
        c = wmma4(a, b, c);
    }
    const float bv = bias[n0 + lr];
#pragma unroll
    for (int j = 0; j < 8; ++j) {
        int m = m0 + j + 8 * half;
        h0[(size_t)m * Dd + n0 + lr] = fmaxf(c[j] + bv, 0.0f);
    }
}

// ---------------------------------------------------------------------------
// K4: logits = h0(32x512) @ w(V x 512)^T + bvec   — WMMA f32 16x16x4
// One wave per 16 vocab columns; computes BOTH 16-row M tiles so the weight
// matrix is streamed exactly once. grid = ceil(V/16), block = 32.
// ---------------------------------------------------------------------------
__global__ __launch_bounds__(32)
void rgcn_cls_gemm(const float* __restrict__ h0,
                   const float* __restrict__ w,
                   const float* __restrict__ bvec,
                   float* __restrict__ out, int V) {
    const int lane = threadIdx.x;
    const int half = lane >> 4;
    const int lr   = lane & 15;
    const int n0   = blockIdx.x * 16;
    const int v    = n0 + lr;
    const bool ok  = (v < V);
    const size_t wrow = (size_t)(ok ? v : 0) * Dd;
    v8f c0 = {0.f, 0.f, 0.f, 0.f, 0.f, 0.f, 0.f, 0.f};
    v8f c1 = {0.f, 0.f, 0.f, 0.f, 0.f, 0.f, 0.f, 0.f};
    for (int k0 = 0; k0 < Dd; k0 += 4) {
        const int ka = k0 + 2 * half;
        v2f b;
        b.x = ok ? w[wrow + ka]     : 0.0f;   // B[k][n] = w[n][k]
        b.y = ok ? w[wrow + ka + 1] : 0.0f;
        v2f a0, a1;
        a0.x = h0[(size_t)lr * Dd + ka];
        a0.y = h0[(size_t)lr * Dd + ka + 1];
        a1.x = h0[(size_t)(16 + lr) * Dd + ka];
        a1.y = h0[(size_t)(16 + lr) * Dd + ka + 1];
        c0 = wmma4(a0, b, c0);
        c1 = wmma4(a1, b, c1);
    }
    if (ok) {
        const float bb = bvec[v];
#pragma unroll
        for (int j = 0; j < 8; ++j) {
            out[(size_t)(j + 8 * half) * V + v]      = c0[j] + bb;
            out[(size_t)(16 + j + 8 * half) * V + v] = c1[j] + bb;
        }
    }
}

// ---------------------------------------------------------------------------
// K5: in-place row-wise log_softmax (one block per row)
// ---------------------------------------------------------------------------
__global__ __launch_bounds__(256)
void rgcn_log_softmax(float* __restrict__ p, int V) {
    __shared__ float sdata[256];
    const int tid = threadIdx.x;
    float* rp = p + (size_t)blockIdx.x * V;

    float m = -INFINITY;
    for (int i = tid; i < V; i += 256) m = fmaxf(m, rp[i]);
    sdata[tid] = m;
    __syncthreads();
    for (int s = 128; s > 0; s >>= 1) {
        if (tid < s) sdata[tid] = fmaxf(sdata[tid], sdata[tid + s]);
        __syncthreads();
    }
    const float rowmax = sdata[0];
    __syncthreads();

    float acc = 0.0f;
    for (int i = tid; i < V; i += 256) acc += expf(rp[i] - rowmax);
    sdata[tid] = acc;
    __syncthreads();
    for (int s = 128; s > 0; s >>= 1) {
        if (tid < s) sdata[tid] += sdata[tid + s];
        __syncthreads();
    }
    const float lse = logf(sdata[0]);
    __syncthreads();

    for (int i = tid; i < V; i += 256) rp[i] = rp[i] - rowmax - lse;
}

// ---------------------------------------------------------------------------
extern "C" void kernel_launch(void* const* d_in, const int* in_sizes, int n_in,
                              void* d_out, int out_size, void* d_ws, size_t ws_size,
                              hipStream_t stream) {
    const float* x        = (const float*)d_in[0];   // (B, N, D)
    const int*   ei       = (const int*)  d_in[1];   // (B, 2, E)
    const int*   et       = (const int*)  d_in[2];   // (B, E)
    const float* basis    = (const float*)d_in[3];   // (R, D, D)
    const float* comp     = (const float*)d_in[4];   // (R, R)
    const float* root     = (const float*)d_in[5];   // (D, D)
    const float* biasconv = (const float*)d_in[6];   // (D,)
    const float* w_global = (const float*)d_in[7];   // (VG, D)
    const float* b_global = (const float*)d_in[8];   // (VG,)
    const float* w_sense  = (const float*)d_in[9];   // (VS, D)
    const float* b_sense  = (const float*)d_in[10];  // (VS,)

    float* out = (float*)d_out;                      // [32*VG | 32*VS]

    // Workspace layout (floats)
    float* ws    = (float*)d_ws;
    float* Wall  = ws;                               // (R+1)*D*D = 2,359,296
    float* cnt0  = Wall + (size_t)(Rr + 1) * Dd * Dd;// B*R       = 256
    float* agg0  = cnt0 + Bsz * Rr;                  // B*R*D     = 131,072
    float* Aext  = agg0 + (size_t)Bsz * Rr * Dd;     // B*KX      = 147,456
    float* h0    = Aext + (size_t)Bsz * KX;          // B*D       = 16,384

    // K0: relation weights (+ root slab)
    {
        int total = (Rr + 1) * Dd * Dd;
        rgcn_build_weights<<<(total + 255) / 256, 256, 0, stream>>>(root, comp, basis, Wall);
    }
    // K1: deterministic node-0 neighbor aggregation
    rgcn_edge_agg<<<Bsz, 256, 0, stream>>>(ei, et, x, agg0, cnt0);
    // K2: extended activations
    {
        int total = Bsz * KX;
        rgcn_build_aext<<<(total + 255) / 256, 256, 0, stream>>>(x, agg0, cnt0, Aext);
    }
    // K3: conv GEMM -> h0 (WMMA)
    {
        dim3 grid(Dd / 16, Bsz / 16);
        rgcn_conv_gemm<<<grid, 32, 0, stream>>>(Aext, Wall, biasconv, h0);
    }
    // K4: classifier GEMMs (WMMA)
    rgcn_cls_gemm<<<(VG + 15) / 16, 32, 0, stream>>>(h0, w_global, b_global, out, VG);
    rgcn_cls_gemm<<<(VS + 15) / 16, 32, 0, stream>>>(h0, w_sense, b_sense,
                                                     out + (size_t)Bsz * VG, VS);
    // K5: log_softmax rows
    rgcn_log_softmax<<<Bsz, 256, 0, stream>>>(out, VG);
    rgcn_log_softmax<<<Bsz, 256, 0, stream>>>(out + (size_t)Bsz * VG, VS);
}